// SLDS_72129680769403
// MI455X (gfx1250) — compile-verified
//
#include <hip/hip_runtime.h>
#include <math.h>

#define HID 64
#define OBS 32
#define KC  32
#define SLD 33   // padded ld for S to dodge LDS bank conflicts

typedef __attribute__((ext_vector_type(2))) float v2f;
typedef __attribute__((ext_vector_type(8))) float v8f;

// Cooperative 8-wave GEMM using V_WMMA_F32_16X16X4_F32.
// D(Mtiles*16 x Ntiles*16) = Am(row-major, lda) * Bm (row-major ldb, or Bm^T if btrans)
// Fragment layouts per CDNA5 ISA 7.12.2:
//   A 16x4 f32 : lanes0-15 K=0/1 in v0/v1 ; lanes16-31 K=2/3
//   B 4x16 f32 : VGPR v, lanes0-15 -> K=v, lanes16-31 -> K=v+2, N=lane%16
//   C/D 16x16  : VGPR v, lanes0-15 -> M=v, lanes16-31 -> M=v+8, N=lane%16
__device__ __forceinline__ void wmma_gemm(const float* Am, int lda,
                                          const float* Bm, int ldb, int btrans,
                                          float* Dm, int ldd,
                                          int Mtiles, int Ntiles, int Kchunks,
                                          int wave, int half, int r)
{
    const int nT = Mtiles * Ntiles;
    for (int tile = wave; tile < nT; tile += 8) {   // uniform per-wave -> EXEC stays all-ones
        const int mt = tile / Ntiles;
        const int nt = tile - mt * Ntiles;
        v8f acc = {0.f,0.f,0.f,0.f,0.f,0.f,0.f,0.f};
        for (int kk = 0; kk < Kchunks; ++kk) {
            const int k0 = kk * 4 + 2 * half;
            v2f a, b;
            a.x = Am[(mt*16 + r)*lda + k0];
            a.y = Am[(mt*16 + r)*lda + k0 + 1];
            if (btrans) {
                b.x = Bm[(nt*16 + r)*ldb + k0];
                b.y = Bm[(nt*16 + r)*ldb + k0 + 1];
            } else {
                b.x = Bm[(k0    )*ldb + nt*16 + r];
                b.y = Bm[(k0 + 1)*ldb + nt*16 + r];
            }
            acc = __builtin_amdgcn_wmma_f32_16x16x4_f32(false, a, false, b,
                                                        (short)0, acc, false, false);
        }
#pragma unroll
        for (int vv = 0; vv < 8; ++vv)
            Dm[(mt*16 + vv + 8*half)*ldd + nt*16 + r] = acc[vv];
    }
}

// trans_log = log_softmax(transition_logits, axis=-1); one block per row
__global__ void slds_prep(const float* __restrict__ logits, float* __restrict__ transLog)
{
    const int row = blockIdx.x, t = threadIdx.x;
    __shared__ float tmp[KC];
    __shared__ float lse;
    float v = logits[row*KC + t];
    tmp[t] = v;
    __syncthreads();
    if (t == 0) {
        float mx = -1e30f;
        for (int k = 0; k < KC; ++k) mx = fmaxf(mx, tmp[k]);
        float s = 0.f;
        for (int k = 0; k < KC; ++k) s += __expf(tmp[k] - mx);
        lse = mx + __logf(s);
    }
    __syncthreads();
    transLog[row*KC + t] = v - lse;
}

// One fused SLDS step: moment-matched predict + Kalman observation update.
// grid = 32 blocks (one per component j), 256 threads (8 waves).
__global__ __launch_bounds__(256) void slds_step(
    const float* __restrict__ transLog,   // (K,K) log_softmax rows
    const float* __restrict__ A,          // (K,H,H)
    const float* __restrict__ logQ,       // (H,)
    const float* __restrict__ C,          // (H,D)
    const float* __restrict__ logR,       // (D,)
    const float* __restrict__ y,          // (D,) observation at this step
    const float* __restrict__ lw_in,
    const float* __restrict__ mu_in,      // (K,H)
    const float* __restrict__ P_in,       // (K,H,H)
    float* __restrict__ lw_out,
    float* __restrict__ mu_out,
    float* __restrict__ P_out,
    int init)
{
    __shared__ float sM [HID*HID];   // M_j, later E = Kg*PC^T
    __shared__ float sT [HID*HID];   // T = A*M, later PC (64x32) + Kg (64x32)
    __shared__ float sPp[HID*HID];   // predicted covariance
    __shared__ float sS [OBS*SLD];   // innovation cov -> Cholesky factor L
    __shared__ float sr [KC];
    __shared__ float mubar [HID];
    __shared__ float mupred[HID];
    __shared__ float vv_[OBS];
    __shared__ float av_[OBS];
    __shared__ float sc[2];          // [0]=predicted log weight, [1]=logdet(S)

    const int tid  = threadIdx.x;
    const int wave = tid >> 5, lane = tid & 31, half = lane >> 4, r = lane & 15;
    const int j = blockIdx.x;
    const float* Aj = A + (size_t)j*HID*HID;
    float* PC = sT;                  // 64x32
    float* Kg = sT + HID*OBS;        // 64x32

    // ---------------- predict (moment matching collapsed over k) ----------------
    if (!init) {
        if (tid == 0) {
            float row[KC];
            float mx = -1e30f;
            for (int k = 0; k < KC; ++k) {
                row[k] = transLog[k*KC + j] + lw_in[k];
                mx = fmaxf(mx, row[k]);
            }
            float s = 0.f;
            for (int k = 0; k < KC; ++k) s += __expf(row[k] - mx);
            float lse = mx + __logf(s);
            sc[0] = lse;
            for (int k = 0; k < KC; ++k) sr[k] = __expf(row[k] - lse);
        }
        __syncthreads();
        if (tid < HID) {
            float s = 0.f;
            for (int k = 0; k < KC; ++k) s += sr[k] * mu_in[k*HID + tid];
            mubar[tid] = s;
        }
        // M_j = sum_k r_k (P_k + mu_k mu_k^T)  -- second moment mixture
        for (int idx = tid; idx < HID*HID; idx += 256) {
            const int h = idx >> 6, g = idx & 63;
            float s = 0.f;
            for (int k = 0; k < KC; ++k)
                s += sr[k] * (P_in[(size_t)k*HID*HID + idx] + mu_in[k*HID+h]*mu_in[k*HID+g]);
            sM[idx] = s;
        }
        __syncthreads();
        wmma_gemm(Aj, HID, sM, HID, 0, sT, HID, 4, 4, 16, wave, half, r);   // T = A*M
        __syncthreads();
        wmma_gemm(sT, HID, Aj, HID, 1, sPp, HID, 4, 4, 16, wave, half, r);  // Pp = T*A^T
        if (tid < HID) {
            float s = 0.f;
            for (int g = 0; g < HID; ++g) s += Aj[tid*HID + g] * mubar[g];
            mupred[tid] = s;
        }
        __syncthreads();
        // Pp += diag(Q) - mupred mupred^T
        for (int idx = tid; idx < HID*HID; idx += 256) {
            const int h = idx >> 6, g = idx & 63;
            float val = sPp[idx] - mupred[h]*mupred[g];
            if (h == g) val += __expf(logQ[h]);
            sPp[idx] = val;
        }
        __syncthreads();
        // symmetrize in place (each unordered pair touched by exactly one thread)
        for (int idx = tid; idx < HID*HID; idx += 256) {
            const int h = idx >> 6, g = idx & 63;
            if (h < g) {
                float m = 0.5f*(sPp[h*HID+g] + sPp[g*HID+h]);
                sPp[h*HID+g] = m; sPp[g*HID+h] = m;
            }
        }
        __syncthreads();
    } else {
        // t == 0: lw = trans_log[0], mu = 0, P = I
        if (tid == 0) sc[0] = transLog[j];
        if (tid < HID) mupred[tid] = 0.f;
        for (int idx = tid; idx < HID*HID; idx += 256) {
            const int h = idx >> 6, g = idx & 63;
            sPp[idx] = (h == g) ? 1.f : 0.f;
        }
        __syncthreads();
    }

    // ---------------- Kalman observation update ----------------
    wmma_gemm(sPp, HID, C, OBS, 0, PC, OBS, 4, 2, 16, wave, half, r);       // PC = Pp*C
    __syncthreads();
    // S = C^T * PC + diag(R)
    for (int idx = tid; idx < OBS*OBS; idx += 256) {
        const int d = idx >> 5, e = idx & 31;
        float s = 0.f;
        for (int h = 0; h < HID; ++h) s += C[h*OBS + d] * PC[h*OBS + e];
        if (d == e) s += __expf(logR[d]);
        sS[d*SLD + e] = s;
    }
    // innovation v = y - mupred @ C
    if (tid < OBS) {
        float s = y[tid];
        for (int h = 0; h < HID; ++h) s -= mupred[h] * C[h*OBS + tid];
        vv_[tid] = s;
    }
    __syncthreads();

    // Cholesky S = L L^T (lower triangle of sS)
    for (int kc = 0; kc < OBS; ++kc) {
        if (tid == 0) sS[kc*SLD + kc] = sqrtf(sS[kc*SLD + kc]);
        __syncthreads();
        if (tid > kc && tid < OBS) sS[tid*SLD + kc] /= sS[kc*SLD + kc];
        __syncthreads();
        for (int idx = tid; idx < OBS*OBS; idx += 256) {
            const int i = idx >> 5, jj = idx & 31;
            if (i > kc && jj > kc && jj <= i)
                sS[i*SLD + jj] -= sS[i*SLD + kc] * sS[jj*SLD + kc];
        }
        __syncthreads();
    }
    if (tid == 0) {
        float ld = 0.f;
        for (int i = 0; i < OBS; ++i) ld += __logf(sS[i*SLD + i]);
        sc[1] = 2.f * ld;
    }
    __syncthreads();

    // Triangular solves: Kg[h,:] = S^{-1} PC[h,:]^T (threads 0..63), a = S^{-1} v (thread 64)
    if (tid < HID) {
        float yv[OBS];
        const float* b = &PC[tid*OBS];
        for (int i = 0; i < OBS; ++i) {
            float s = b[i];
            for (int jj = 0; jj < i; ++jj) s -= sS[i*SLD + jj] * yv[jj];
            yv[i] = s / sS[i*SLD + i];
        }
        for (int i = OBS-1; i >= 0; --i) {
            float s = yv[i];
            for (int jj = i+1; jj < OBS; ++jj) s -= sS[jj*SLD + i] * yv[jj];
            yv[i] = s / sS[i*SLD + i];
        }
        for (int i = 0; i < OBS; ++i) Kg[tid*OBS + i] = yv[i];
    } else if (tid == HID) {
        float yv[OBS];
        for (int i = 0; i < OBS; ++i) {
            float s = vv_[i];
            for (int jj = 0; jj < i; ++jj) s -= sS[i*SLD + jj] * yv[jj];
            yv[i] = s / sS[i*SLD + i];
        }
        for (int i = OBS-1; i >= 0; --i) {
            float s = yv[i];
            for (int jj = i+1; jj < OBS; ++jj) s -= sS[jj*SLD + i] * yv[jj];
            yv[i] = s / sS[i*SLD + i];
        }
        for (int i = 0; i < OBS; ++i) av_[i] = yv[i];
    }
    __syncthreads();

    if (tid == 0) {
        float q = 0.f;
        for (int d = 0; d < OBS; ++d) q += vv_[d] * av_[d];
        const float LOG2PI = 1.8378770664093453f;
        lw_out[j] = sc[0] - 0.5f * ((float)OBS * LOG2PI + sc[1] + q);
    }
    if (tid < HID) {
        float s = mupred[tid];
        for (int d = 0; d < OBS; ++d) s += Kg[tid*OBS + d] * vv_[d];
        mu_out[(size_t)j*HID + tid] = s;
    }

    // E = Kg * PC^T  (uniform control flow; inputs complete at last barrier)
    wmma_gemm(Kg, OBS, PC, OBS, 1, sM, HID, 4, 4, 8, wave, half, r);
    __syncthreads();
    // P_out = 0.5*((Pp - E) + (Pp - E)^T)
    for (int idx = tid; idx < HID*HID; idx += 256) {
        const int h = idx >> 6, g = idx & 63;
        float val = 0.5f * ((sPp[h*HID+g] - sM[h*HID+g]) + (sPp[g*HID+h] - sM[g*HID+h]));
        P_out[(size_t)j*HID*HID + idx] = val;
    }
}

__global__ void slds_final(const float* __restrict__ lw, float* __restrict__ out)
{
    __shared__ float tmp[KC];
    const int t = threadIdx.x;
    tmp[t] = lw[t];
    __syncthreads();
    if (t == 0) {
        float mx = -1e30f;
        for (int k = 0; k < KC; ++k) mx = fmaxf(mx, tmp[k]);
        float s = 0.f;
        for (int k = 0; k < KC; ++k) s += __expf(tmp[k] - mx);
        out[0] = mx + __logf(s);
    }
}

extern "C" void kernel_launch(void* const* d_in, const int* in_sizes, int n_in,
                              void* d_out, int out_size, void* d_ws, size_t ws_size,
                              hipStream_t stream)
{
    const float* data   = (const float*)d_in[0];  // (T, OBS)
    const float* tlogit = (const float*)d_in[1];  // (K, K)
    const float* A      = (const float*)d_in[2];  // (K, H, H)
    const float* logQ   = (const float*)d_in[3];  // (H,)
    const float* C      = (const float*)d_in[4];  // (H, D)
    const float* logR   = (const float*)d_in[5];  // (D,)

    const int T = in_sizes[0] / OBS;              // 500

    float* ws = (float*)d_ws;
    float* transLog = ws;                                         // 1024 floats
    const size_t BUF = 64 + (size_t)KC*HID + (size_t)KC*HID*HID;  // lw | mu | P
    float* b0 = ws + 1024;
    float* b1 = b0 + BUF;

    slds_prep<<<KC, KC, 0, stream>>>(tlogit, transLog);

    for (int t = 0; t < T; ++t) {
        float* bin  = (t & 1) ? b1 : b0;
        float* bout = (t & 1) ? b0 : b1;
        slds_step<<<KC, 256, 0, stream>>>(
            transLog, A, logQ, C, logR,
            data + (size_t)t * OBS,
            bin,  bin  + 64, bin  + 64 + (size_t)KC*HID,
            bout, bout + 64, bout + 64 + (size_t)KC*HID,
            (t == 0) ? 1 : 0);
    }

    float* bfin = (T & 1) ? b0 : b1;  // after step t, result is in bout of that t
    // t = T-1: bout = (T-1)&1 ? b0 : b1  -> T even => bout = b0
    bfin = ((T - 1) & 1) ? b0 : b1;
    slds_final<<<1, KC, 0, stream>>>(bfin, (float*)d_out);
}